// MessagePassingConvolution_19172734009580
// MI455X (gfx1250) — compile-verified
//
#include <hip/hip_runtime.h>

typedef __attribute__((ext_vector_type(2))) float v2f;
typedef __attribute__((ext_vector_type(8))) float v8f;

#define N_NODES    16384
#define N_EDGES    131072
#define MUL        64
#define N_RADIAL   8
#define HIDDEN     64
#define NUM_IRREPS 384
#define TILE_E     16

// padded LDS strides (floats) to avoid 16-way bank conflicts on A-frag reads
#define R_STRIDE 12
#define H_STRIDE 65
#define W_STRIDE 385

// silu via hardware reciprocal (v_rcp_f32) instead of IEEE divide chain
__device__ __forceinline__ float silu_f(float x) {
    return x * __builtin_amdgcn_rcpf(1.0f + __expf(-x));
}

__global__ __launch_bounds__(256) void zero_out_kernel(float* __restrict__ out, int n4) {
    int i = blockIdx.x * blockDim.x + threadIdx.x;
    if (i < n4) {
        *(float4*)(out + 4 * (size_t)i) = make_float4(0.f, 0.f, 0.f, 0.f);
    }
}

__global__ __launch_bounds__(256) void edge_mlp_tp_scatter(
    const float* __restrict__ node_feats,   // N x 256  (s[64] | v[64][3])
    const float* __restrict__ edge_features,// E x 4    (es | ev[3])
    const float* __restrict__ radial,       // E x 8
    const float* __restrict__ w1,           // 8  x 64
    const float* __restrict__ w2,           // 64 x 64
    const float* __restrict__ w3,           // 64 x 384
    const int*   __restrict__ senders,      // E
    const int*   __restrict__ receivers,    // E
    float*       __restrict__ out)          // N x 768  (s[192] | v[192][3])
{
    __shared__ float r_lds[TILE_E * R_STRIDE];
    __shared__ float h1_lds[TILE_E * H_STRIDE];
    __shared__ float h2_lds[TILE_E * H_STRIDE];
    __shared__ float w_lds[TILE_E * W_STRIDE];
    __shared__ int   snd_lds[TILE_E];
    __shared__ int   rcv_lds[TILE_E];
    __shared__ float ef_lds[TILE_E][4];

    const int tid  = threadIdx.x;
    const int e0   = blockIdx.x * TILE_E;
    const int lane = tid & 31;
    const int wv   = tid >> 5;     // wave 0..7
    const int half = lane >> 4;    // 0/1
    const int lm   = lane & 15;    // 0..15

    // ---- stage A: edge tile metadata + radial features -> LDS ----
    if (tid < TILE_E * N_RADIAL) {          // 128 threads
        int er = tid >> 3;
        int k  = tid & 7;
        r_lds[er * R_STRIDE + k] = radial[(size_t)(e0 + er) * N_RADIAL + k];
    }
    if (tid < TILE_E) {
        snd_lds[tid] = senders[e0 + tid];
        rcv_lds[tid] = receivers[e0 + tid];
    }
    if (tid >= 64 && tid < 64 + TILE_E * 4) {
        int t = tid - 64;
        ef_lds[t >> 2][t & 3] = edge_features[(size_t)(e0 + (t >> 2)) * 4 + (t & 3)];
    }
    __syncthreads();

    const float rs8    = 0.35355339059327373f;  // 1/sqrt(8)  (layer1 norm)
    const float inv8   = 0.125f;                // 1/sqrt(64)
    const float wscale = inv8 * rs8;            // layer3 scale * 1/sqrt(AVG_NUM_NEIGHBORS)

    // ---- stage B: layer 1  h1 = silu( r(16x8) @ w1(8x64) / sqrt(8) ), waves 0..3 ----
    if (wv < 4) {
        const int n0 = wv * 16;
        v8f c = {};
        #pragma unroll
        for (int k = 0; k < N_RADIAL; k += 4) {
            v2f a, b;
            a[0] = r_lds[lm * R_STRIDE + k + 2 * half + 0];
            a[1] = r_lds[lm * R_STRIDE + k + 2 * half + 1];
            b[0] = w1[(k + 2 * half + 0) * HIDDEN + n0 + lm];
            b[1] = w1[(k + 2 * half + 1) * HIDDEN + n0 + lm];
            c = __builtin_amdgcn_wmma_f32_16x16x4_f32(false, a, false, b,
                                                      (short)0, c, false, false);
        }
        #pragma unroll
        for (int i = 0; i < 8; ++i) {
            int m = i + 8 * half;             // edge row
            h1_lds[m * H_STRIDE + n0 + lm] = silu_f(c[i] * rs8);
        }
    }
    __syncthreads();

    // ---- stage C: layer 2  h2 = silu( h1(16x64) @ w2(64x64) / 8 ), waves 0..3 ----
    if (wv < 4) {
        const int n0 = wv * 16;
        v8f c = {};
        #pragma unroll 4
        for (int k = 0; k < HIDDEN; k += 4) {
            v2f a, b;
            a[0] = h1_lds[lm * H_STRIDE + k + 2 * half + 0];
            a[1] = h1_lds[lm * H_STRIDE + k + 2 * half + 1];
            b[0] = w2[(k + 2 * half + 0) * HIDDEN + n0 + lm];
            b[1] = w2[(k + 2 * half + 1) * HIDDEN + n0 + lm];
            c = __builtin_amdgcn_wmma_f32_16x16x4_f32(false, a, false, b,
                                                      (short)0, c, false, false);
        }
        #pragma unroll
        for (int i = 0; i < 8; ++i) {
            int m = i + 8 * half;
            h2_lds[m * H_STRIDE + n0 + lm] = silu_f(c[i] * inv8);
        }
    }
    __syncthreads();

    // ---- stage D: layer 3  w = h2(16x64) @ w3(64x384) * wscale, all 8 waves x 3 tiles ----
    #pragma unroll
    for (int t = 0; t < 3; ++t) {
        const int n0 = (wv * 3 + t) * 16;
        v8f c = {};
        #pragma unroll 4
        for (int k = 0; k < HIDDEN; k += 4) {
            v2f a, b;
            a[0] = h2_lds[lm * H_STRIDE + k + 2 * half + 0];
            a[1] = h2_lds[lm * H_STRIDE + k + 2 * half + 1];
            b[0] = w3[(k + 2 * half + 0) * NUM_IRREPS + n0 + lm];
            b[1] = w3[(k + 2 * half + 1) * NUM_IRREPS + n0 + lm];
            c = __builtin_amdgcn_wmma_f32_16x16x4_f32(false, a, false, b,
                                                      (short)0, c, false, false);
        }
        #pragma unroll
        for (int i = 0; i < 8; ++i) {
            int m = i + 8 * half;
            w_lds[m * W_STRIDE + n0 + lm] = c[i] * wscale;
        }
    }
    __syncthreads();

    // ---- stage E: tensor product + weighted scatter-add ----
    // 16 threads per edge, 4 mul-channels per thread
    const int el = tid >> 4;    // edge within tile
    const int q  = tid & 15;
    const int m0 = q * 4;

    const int   s_node = snd_lds[el];
    const int   r_node = rcv_lds[el];
    const float es  = ef_lds[el][0];
    const float ev0 = ef_lds[el][1];
    const float ev1 = ef_lds[el][2];
    const float ev2 = ef_lds[el][3];
    const float* nf   = node_feats + (size_t)s_node * (MUL + 3 * MUL);
    const float* wrow = &w_lds[el * W_STRIDE];
    float*       ob   = out + (size_t)r_node * 768;

    // gather sender features: s[m0..m0+3], v[m0..m0+3][0..2] (contiguous, 16B aligned)
    const float4 sv = *(const float4*)(nf + m0);
    const float4 va = *(const float4*)(nf + MUL + m0 * 3);
    const float4 vb = *(const float4*)(nf + MUL + m0 * 3 + 4);
    const float4 vc = *(const float4*)(nf + MUL + m0 * 3 + 8);
    const float sbuf[4]  = {sv.x, sv.y, sv.z, sv.w};
    const float vbuf[12] = {va.x, va.y, va.z, va.w,
                            vb.x, vb.y, vb.z, vb.w,
                            vc.x, vc.y, vc.z, vc.w};
    const float inv3 = 0.57735026918962576f;  // 1/sqrt(3)

    #pragma unroll
    for (int j = 0; j < 4; ++j) {
        const int   m  = m0 + j;
        const float si = sbuf[j];
        const float v0 = vbuf[j * 3 + 0];
        const float v1 = vbuf[j * 3 + 1];
        const float v2 = vbuf[j * 3 + 2];
        const float dot = (v0 * ev0 + v1 * ev1 + v2 * ev2) * inv3;

        // scalar channels: [s_e | s_e*es | (v.ev)/sqrt3] * w[:192]
        atomicAdd(ob + m,        si       * wrow[m]);
        atomicAdd(ob + 64 + m,   si * es  * wrow[64 + m]);
        atomicAdd(ob + 128 + m,  dot      * wrow[128 + m]);

        // vector channels: [v_e | s_e*ev | v_e*es] * w[192:384]
        const float wva = wrow[192 + m];
        const float wvb = wrow[256 + m];
        const float wvc = wrow[320 + m];
        const int b1 = 192 + m * 3;
        atomicAdd(ob + b1 + 0, v0 * wva);
        atomicAdd(ob + b1 + 1, v1 * wva);
        atomicAdd(ob + b1 + 2, v2 * wva);
        const int b2 = 192 + (64 + m) * 3;
        atomicAdd(ob + b2 + 0, si * ev0 * wvb);
        atomicAdd(ob + b2 + 1, si * ev1 * wvb);
        atomicAdd(ob + b2 + 2, si * ev2 * wvb);
        const int b3 = 192 + (128 + m) * 3;
        atomicAdd(ob + b3 + 0, v0 * es * wvc);
        atomicAdd(ob + b3 + 1, v1 * es * wvc);
        atomicAdd(ob + b3 + 2, v2 * es * wvc);
    }
}

extern "C" void kernel_launch(void* const* d_in, const int* in_sizes, int n_in,
                              void* d_out, int out_size, void* d_ws, size_t ws_size,
                              hipStream_t stream) {
    const float* node_feats    = (const float*)d_in[0];
    const float* edge_features = (const float*)d_in[1];
    const float* radial        = (const float*)d_in[2];
    const float* w1            = (const float*)d_in[3];
    const float* w2            = (const float*)d_in[4];
    const float* w3            = (const float*)d_in[5];
    const int*   senders       = (const int*)d_in[6];
    const int*   receivers     = (const int*)d_in[7];
    float*       out           = (float*)d_out;

    // out_size = 16384*768, divisible by 4
    const int n4 = out_size / 4;
    zero_out_kernel<<<(n4 + 255) / 256, 256, 0, stream>>>(out, n4);

    edge_mlp_tp_scatter<<<N_EDGES / TILE_E, 256, 0, stream>>>(
        node_feats, edge_features, radial, w1, w2, w3, senders, receivers, out);
}